// Model_20959440404502
// MI455X (gfx1250) — compile-verified
//
#include <hip/hip_runtime.h>

typedef __attribute__((ext_vector_type(2))) float v2f;
typedef __attribute__((ext_vector_type(8))) float v8f;

#define B_DIM 64
#define L_DIM 8192
#define C_DIM 256
#define NSEG  32
#define LSEG  (L_DIM / NSEG)   // 256
#define TILES (LSEG / 16)      // 16 tiles of 16 rows per segment

// ---------------------------------------------------------------------------
// K1: per-segment column sums.  thread = (b, s, c); writes ws[b*NSEG*C + s*C + c]
// ---------------------------------------------------------------------------
__global__ void seg_sum_kernel(const float* __restrict__ x, float* __restrict__ ws) {
    int tid = blockIdx.x * blockDim.x + threadIdx.x;      // B*NSEG*C threads
    int c = tid & (C_DIM - 1);
    int s = (tid >> 8) & (NSEG - 1);
    int b = tid >> 13;
    const float* p = x + ((size_t)(b * L_DIM + s * LSEG) * C_DIM + c);
    float a0 = 0.f, a1 = 0.f, a2 = 0.f, a3 = 0.f;
    #pragma unroll 4
    for (int l = 0; l < LSEG; l += 4) {
        a0 += p[(size_t)(l + 0) * C_DIM];
        a1 += p[(size_t)(l + 1) * C_DIM];
        a2 += p[(size_t)(l + 2) * C_DIM];
        a3 += p[(size_t)(l + 3) * C_DIM];
    }
    ws[tid] = (a0 + a1) + (a2 + a3);
}

// ---------------------------------------------------------------------------
// K2: in-place exclusive scan of segment sums along s.  thread = (b, c)
// ---------------------------------------------------------------------------
__global__ void seg_scan_kernel(float* __restrict__ ws) {
    int tid = blockIdx.x * blockDim.x + threadIdx.x;      // B*C threads
    int c = tid & (C_DIM - 1);
    int b = tid >> 8;
    size_t base = (size_t)b * NSEG * C_DIM + c;
    float run = 0.f;
    for (int s = 0; s < NSEG; ++s) {
        size_t i = base + (size_t)s * C_DIM;
        float t = ws[i];
        ws[i] = run;
        run += t;
    }
}

// ---------------------------------------------------------------------------
// K3: WMMA tile scan.  One wave handles (b, channel-group of 16, segment).
// Per 16x16 tile:  D = T(lower-tri ones) * X + C(carry broadcast), 4 chained
// v_wmma_f32_16x16x4_f32.  New carry = D row 15, fetched via ds_bpermute.
// ---------------------------------------------------------------------------
__global__ void scan_wmma_kernel(const float* __restrict__ x,
                                 float* __restrict__ y,
                                 const float* __restrict__ ws) {
    const int lane    = threadIdx.x & 31;
    const int lane_lo = lane & 15;     // = M for A, = N for B/C/D
    const int lane_hi = lane >> 4;

    int w  = blockIdx.x * (blockDim.x >> 5) + (threadIdx.x >> 5);
    int s  = w & (NSEG - 1);
    int cg = (w >> 5) & 15;
    int b  = w >> 9;
    int c0 = cg * 16;

    // A = 16x16 lower-triangular ones, split into 4 slices of 16x4.
    // ISA layout (32-bit A 16x4): lane holds M = lane_lo; VGPR j holds
    // K = 2*lane_hi + j.  Slice k covers absolute K = 4k .. 4k+3.
    v2f a[4];
    #pragma unroll
    for (int k = 0; k < 4; ++k) {
        int kk = 4 * k + 2 * lane_hi;
        a[k].x = (kk     <= lane_lo) ? 1.0f : 0.0f;
        a[k].y = (kk + 1 <= lane_lo) ? 1.0f : 0.0f;
    }

    // Running carry for this wave's 16 channels (exclusive segment prefix).
    float carry = ws[((size_t)b * NSEG + s) * C_DIM + c0 + lane_lo];

    const size_t rs = C_DIM;  // row stride in floats
    const float* xin  = x + (size_t)(b * L_DIM + s * LSEG) * C_DIM + c0 + lane_lo;
    float*       yout = y + (size_t)(b * L_DIM + s * LSEG) * C_DIM + c0 + lane_lo;

    for (int t = 0; t < TILES; ++t) {
        const float* xt = xin + (size_t)t * 16 * rs;
        // B layout (4x16 per slice): VGPR j holds row K = 2*lane_hi + j,
        // column N = lane_lo.
        v2f bm[4];
        #pragma unroll
        for (int k = 0; k < 4; ++k) {
            int r0 = 4 * k + 2 * lane_hi;
            bm[k].x = xt[(size_t)(r0    ) * rs];
            bm[k].y = xt[(size_t)(r0 + 1) * rs];
        }

        // C = carry broadcast into all 16 rows of the accumulator.
        v8f c;
        #pragma unroll
        for (int i = 0; i < 8; ++i) c[i] = carry;

        #pragma unroll
        for (int k = 0; k < 4; ++k)
            c = __builtin_amdgcn_wmma_f32_16x16x4_f32(
                    false, a[k], false, bm[k], (short)0, c, false, false);

        // D layout: VGPR r holds row M = r + 8*lane_hi, column N = lane_lo.
        float* yt = yout + (size_t)t * 16 * rs;
        #pragma unroll
        for (int r = 0; r < 8; ++r)
            yt[(size_t)(r + 8 * lane_hi) * rs] = c[r];

        // carry = D[15][n] = VGPR7 of lane (16+n); broadcast to all lanes.
        int ci = __builtin_amdgcn_ds_bpermute((16 + lane_lo) * 4,
                                              __builtin_bit_cast(int, c[7]));
        carry = __builtin_bit_cast(float, ci);
    }
}

// ---------------------------------------------------------------------------
// Fallback (no workspace): one thread per (b,c) line, serial scan.
// ---------------------------------------------------------------------------
__global__ void serial_scan_kernel(const float* __restrict__ x,
                                   float* __restrict__ y) {
    int tid = blockIdx.x * blockDim.x + threadIdx.x;      // B*C threads
    int c = tid & (C_DIM - 1);
    int b = tid >> 8;
    const float* p = x + (size_t)b * L_DIM * C_DIM + c;
    float*       q = y + (size_t)b * L_DIM * C_DIM + c;
    float acc = 0.f;
    for (int l = 0; l < L_DIM; ++l) {
        acc += p[(size_t)l * C_DIM];
        q[(size_t)l * C_DIM] = acc;
    }
}

extern "C" void kernel_launch(void* const* d_in, const int* in_sizes, int n_in,
                              void* d_out, int out_size, void* d_ws, size_t ws_size,
                              hipStream_t stream) {
    const float* x = (const float*)d_in[0];
    float*       y = (float*)d_out;
    const size_t ws_need = (size_t)B_DIM * NSEG * C_DIM * sizeof(float);  // 2 MB

    if (ws_size >= ws_need) {
        float* ws = (float*)d_ws;
        seg_sum_kernel<<<(B_DIM * NSEG * C_DIM) / 256, 256, 0, stream>>>(x, ws);
        seg_scan_kernel<<<(B_DIM * C_DIM) / 256, 256, 0, stream>>>(ws);
        // 64 b * 16 cgroups * 32 segments = 32768 waves, 8 waves/block
        scan_wmma_kernel<<<(B_DIM * 16 * NSEG) / 8, 256, 0, stream>>>(x, y, ws);
    } else {
        serial_scan_kernel<<<(B_DIM * C_DIM) / 256, 256, 0, stream>>>(x, y);
    }
}